// RobustGCNConv_18047452578193
// MI455X (gfx1250) — compile-verified
//
#include <hip/hip_runtime.h>
#include <cstdint>
#include <cstddef>

// RobustGCNConv for MI455X (gfx1250, wave32).
//
// Phase 1: fused dual GEMM (mean@W_mean, var@W_var) via V_WMMA_F32_16X16X4_F32,
//          + bias + elu/relu + exp(-v) attention gate -> workspace (L2-resident, 51MB).
// Phase 2: zero-init outputs.
// Phase 3: COO scatter: wave per edge, float4 gathers, no-return f32 atomics.

typedef __attribute__((ext_vector_type(2))) float v2f;
typedef __attribute__((ext_vector_type(8))) float v8f;

#define DIN  128
#define DOUT 128
#define EPSV 1e-6f

// ---------------------------------------------------------------------------
// Kernel 1: WMMA projection + activation + gate.
// Block = 256 threads = 8 waves. Block handles 16 node-rows; wave w handles
// output columns [16w, 16w+16). K loop: 128 / 4 = 32 WMMA steps per GEMM.
// ---------------------------------------------------------------------------
__global__ __launch_bounds__(256) void proj_gate_wmma(
    const float* __restrict__ mean, const float* __restrict__ var,
    const float* __restrict__ Wm,   const float* __restrict__ bm,
    const float* __restrict__ Wv,   const float* __restrict__ bv,
    float* __restrict__ m_ws, float* __restrict__ v_ws, int N)
{
    const int lane    = threadIdx.x & 31;
    const int wave    = threadIdx.x >> 5;        // 0..7 -> N-tile
    const int rowBase = blockIdx.x * 16;
    const int colBase = wave * 16;
    const int l15     = lane & 15;
    const int hi      = lane >> 4;               // 0 or 1 (lane half)

    // A-matrix 16x4 f32 layout (ISA 7.12.2): lanes 0-15: VGPR0=K0,VGPR1=K1;
    // lanes 16-31: VGPR0=K2,VGPR1=K3.  Row = lane&15.
    int arow = rowBase + l15;
    if (arow >= N) arow = N - 1;                 // clamp for tail (stores guarded)
    const float* aM = mean + (size_t)arow * DIN + 2 * hi;
    const float* aV = var  + (size_t)arow * DIN + 2 * hi;

    // B-matrix 4x16 f32: VGPR0 = row K=(0|2), VGPR1 = row K=(1|3), N = lane&15.
    const float* bM = Wm + (size_t)(2 * hi) * DOUT + colBase + l15;
    const float* bV = Wv + (size_t)(2 * hi) * DOUT + colBase + l15;

    v8f cm = {};
    v8f cv = {};
#pragma unroll 4
    for (int k = 0; k < DIN; k += 4) {
        v2f am; am.x = aM[k];                 am.y = aM[k + 1];
        v2f av; av.x = aV[k];                 av.y = aV[k + 1];
        v2f wm; wm.x = bM[(size_t)k * DOUT];  wm.y = bM[(size_t)(k + 1) * DOUT];
        v2f wv; wv.x = bV[(size_t)k * DOUT];  wv.y = bV[(size_t)(k + 1) * DOUT];
        cm = __builtin_amdgcn_wmma_f32_16x16x4_f32(false, am, false, wm,
                                                   (short)0, cm, false, false);
        cv = __builtin_amdgcn_wmma_f32_16x16x4_f32(false, av, false, wv,
                                                   (short)0, cv, false, false);
    }

    // C/D layout: VGPR r -> M = r + 8*hi, N = lane&15.
    const int   col   = colBase + l15;
    const float biasM = bm[col];
    const float biasV = bv[col];
#pragma unroll
    for (int r = 0; r < 8; ++r) {
        const int row = rowBase + r + 8 * hi;
        if (row >= N) continue;
        const float xm = cm[r] + biasM;
        const float xv = cv[r] + biasV;
        const float m  = (xm > 0.f) ? xm : (expf(xm) - 1.f);   // elu
        const float v  = ((xv > 0.f) ? xv : 0.f) + EPSV;        // relu + eps
        const float att = expf(-v);
        m_ws[(size_t)row * DOUT + col] = m * att;
        v_ws[(size_t)row * DOUT + col] = v * att * att;
    }
}

// ---------------------------------------------------------------------------
// Kernel 2: zero-init outputs (harness poisons d_out; segment_sum needs 0).
// ---------------------------------------------------------------------------
__global__ __launch_bounds__(256) void zero_f4(float4* __restrict__ p, size_t n4)
{
    size_t i      = (size_t)blockIdx.x * blockDim.x + threadIdx.x;
    size_t stride = (size_t)gridDim.x * blockDim.x;
    const float4 z = make_float4(0.f, 0.f, 0.f, 0.f);
    for (; i < n4; i += stride) p[i] = z;
}

// ---------------------------------------------------------------------------
// Kernel 3: COO scatter-add. One wave (32 lanes) per edge; each lane owns a
// float4 chunk (32*4 = 128 features). Gathers hit L2 (m_ws/v_ws = 51 MB).
// Atomics are no-return global_atomic_add_f32 (STOREcnt path).
// ---------------------------------------------------------------------------
__global__ __launch_bounds__(256) void edge_scatter(
    const float* __restrict__ m_ws, const float* __restrict__ v_ws,
    const float* __restrict__ w0,   const float* __restrict__ w1,
    const int* __restrict__ esrc,   const int* __restrict__ edst,
    float* __restrict__ out_m, float* __restrict__ out_v, int E)
{
    const int lane   = threadIdx.x & 31;
    int       wid    = (int)((blockIdx.x * blockDim.x + threadIdx.x) >> 5);
    const int nwaves = (int)((gridDim.x * blockDim.x) >> 5);

    for (int e = wid; e < E; e += nwaves) {
        const int   s  = esrc[e];
        const int   d  = edst[e];
        const float a0 = w0[e];
        const float a1 = w1[e];

        const float4 m4 = ((const float4*)(m_ws + (size_t)s * DOUT))[lane];
        const float4 v4 = ((const float4*)(v_ws + (size_t)s * DOUT))[lane];

        float* om = out_m + (size_t)d * DOUT + lane * 4;
        float* ov = out_v + (size_t)d * DOUT + lane * 4;
        unsafeAtomicAdd(om + 0, a0 * m4.x);
        unsafeAtomicAdd(om + 1, a0 * m4.y);
        unsafeAtomicAdd(om + 2, a0 * m4.z);
        unsafeAtomicAdd(om + 3, a0 * m4.w);
        unsafeAtomicAdd(ov + 0, a1 * v4.x);
        unsafeAtomicAdd(ov + 1, a1 * v4.y);
        unsafeAtomicAdd(ov + 2, a1 * v4.z);
        unsafeAtomicAdd(ov + 3, a1 * v4.w);
    }
}

// ---------------------------------------------------------------------------
extern "C" void kernel_launch(void* const* d_in, const int* in_sizes, int n_in,
                              void* d_out, int out_size, void* d_ws, size_t ws_size,
                              hipStream_t stream)
{
    const float* mean = (const float*)d_in[0];
    const float* var  = (const float*)d_in[1];
    const float* Wm   = (const float*)d_in[2];
    const float* bm   = (const float*)d_in[3];
    const float* Wv   = (const float*)d_in[4];
    const float* bv   = (const float*)d_in[5];
    const float* w0   = (const float*)d_in[6];
    const float* w1   = (const float*)d_in[7];
    const int*   esrc = (const int*)d_in[8];
    const int*   edst = (const int*)d_in[9];

    const int N = in_sizes[0] / DIN;
    const int E = in_sizes[8];

    float* out_m = (float*)d_out;                       // [N, DOUT]
    float* out_v = out_m + (size_t)N * DOUT;            // [N, DOUT]

    float* m_ws = (float*)d_ws;                         // gated m
    float* v_ws = m_ws + (size_t)N * DOUT;              // gated v
    (void)ws_size; (void)n_in;

    // Phase 1: WMMA projection + gate -> workspace.
    const int nRowBlocks = (N + 15) / 16;
    proj_gate_wmma<<<nRowBlocks, 256, 0, stream>>>(mean, var, Wm, bm, Wv, bv,
                                                   m_ws, v_ws, N);

    // Phase 2: zero outputs (out_size = 2*N*DOUT, divisible by 4).
    const size_t n4 = (size_t)out_size / 4;
    int zblocks = (int)((n4 + 255) / 256);
    if (zblocks > 4096) zblocks = 4096;
    zero_f4<<<zblocks, 256, 0, stream>>>((float4*)d_out, n4);

    // Phase 3: edge scatter (wave per edge, grid-stride).
    edge_scatter<<<2048, 256, 0, stream>>>(m_ws, v_ws, w0, w1, esrc, edst,
                                           out_m, out_v, E);
}